// MyRNNCell_80221399154888
// MI455X (gfx1250) — compile-verified
//
#include <hip/hip_runtime.h>

#ifndef __has_builtin
#define __has_builtin(x) 0
#endif

#if __has_builtin(__builtin_amdgcn_global_load_async_to_lds_b128)
#define HAVE_ASYNC_B128 1
#else
#define HAVE_ASYNC_B128 0
#endif

// 4 x i32 ext-vector, as expected by the async-load builtin, plus
// address-space-qualified pointer types for global (1) and LDS (3).
typedef int v4i __attribute__((ext_vector_type(4)));
typedef __attribute__((address_space(1))) v4i* gptr_v4i;
typedef __attribute__((address_space(3))) v4i* lptr_v4i;

__device__ __forceinline__ void wait_async0() {
#if __has_builtin(__builtin_amdgcn_s_wait_asynccnt)
    __builtin_amdgcn_s_wait_asynccnt(0);
#else
    asm volatile("s_wait_asynccnt 0" ::: "memory");
#endif
}

#define NROW 4096
#define NCOL 4096
#define ROWS_PER_BLOCK 16   // 8 wave32s/block, 2 rows per wave
#define N_STEPS 512

// ---------------------------------------------------------------------------
// Kernel 1: a[row] = sum_j Re(W[row][j]) * x[j]   (f64 accumulate)
// W is complex128 row-major -> one double2 (re, im) per element.
// Bandwidth-bound: streams 256 MB of W exactly once (~11 us at 23.3 TB/s);
// the dead U@z product from the reference is skipped entirely.
// x (16 KB) is async-staged into LDS (ASYNCcnt path), reused by 16 rows.
// Each wave32 owns two rows and keeps 8 independent b128 loads in flight
// per chunk iteration to hide HBM latency.
// ---------------------------------------------------------------------------
__global__ __launch_bounds__(256) void rewx_matvec(const double2* __restrict__ W,
                                                   const float*   __restrict__ x,
                                                   double*        __restrict__ a) {
    __shared__ float xs[NCOL];   // 16 KB
    const int t = threadIdx.x;

#if HAVE_ASYNC_B128
    // Async global->LDS: 4 x b128 per thread = 16 KB total, tracked by ASYNCcnt.
    for (int k = 0; k < 4; ++k) {
        const int idx = (t + k * 256) * 4;       // float index, 16B-aligned chunks
        __builtin_amdgcn_global_load_async_to_lds_b128(
            (gptr_v4i)(x + idx), (lptr_v4i)&xs[idx], 0, 0);
    }
    wait_async0();
#else
    for (int k = 0; k < 4; ++k) {
        const int idx = t + k * 256;
        ((float4*)xs)[idx] = ((const float4*)x)[idx];
    }
#endif
    __syncthreads();

    const int wave = t >> 5;     // wave32
    const int lane = t & 31;
    const int row0 = blockIdx.x * ROWS_PER_BLOCK + wave * 2;

    const double2* __restrict__ wrow0 = W + (size_t)(row0 + 0) * NCOL;
    const double2* __restrict__ wrow1 = W + (size_t)(row0 + 1) * NCOL;

    double acc0 = 0.0;
    double acc1 = 0.0;

    // 32 chunk iterations; each chunk = 128 columns = 2 KB/row for the wave.
    for (int jj = 0; jj < NCOL; jj += 128) {
        // one hoisted prefetch per chunk, ~16 KB ahead in each row stream
        if (jj + 1024 < NCOL) {
            __builtin_prefetch(&wrow0[jj + lane + 1024], 0, 0);  // global_prefetch_b8
            __builtin_prefetch(&wrow1[jj + lane + 1024], 0, 0);
        }
        #pragma unroll
        for (int u = 0; u < 4; ++u) {
            const int j = jj + u * 32 + lane;
            const double2 w0 = wrow0[j];             // global_load_b128
            const double2 w1 = wrow1[j];             // global_load_b128
            const double  xv = (double)xs[j];
            acc0 = fma(w0.x, xv, acc0);
            acc1 = fma(w1.x, xv, acc1);
        }
    }

    // wave32 tree reductions
    for (int off = 16; off > 0; off >>= 1) {
        acc0 += __shfl_down(acc0, off, 32);
        acc1 += __shfl_down(acc1, off, 32);
    }
    if (lane == 0) {
        a[row0 + 0] = acc0;
        a[row0 + 1] = acc1;
    }
}

// ---------------------------------------------------------------------------
// Kernel 2: per-element complex RK4 of dz/dt = (a - |z|^2) * z over [0, 2*pi].
// a is real, b == 1, so the RHS is a real scalar times z. Pure f64 VALU.
// ---------------------------------------------------------------------------
__global__ __launch_bounds__(256) void hopf_rk4(const double* __restrict__ a_vec,
                                                const double* __restrict__ z0,
                                                double*       __restrict__ zT) {
    const int i = blockIdx.x * blockDim.x + threadIdx.x;
    if (i >= NROW) return;

    const double a  = a_vec[i];
    double zr = z0[2 * i + 0];
    double zi = z0[2 * i + 1];

    const double h  = 6.2831853071795864769252867665590 / (double)N_STEPS;
    const double h2 = 0.5 * h;
    const double h6 = h / 6.0;

    #pragma unroll 4
    for (int s = 0; s < N_STEPS; ++s) {
        const double m1  = a - fma(zr, zr, zi * zi);
        const double k1r = m1 * zr, k1i = m1 * zi;

        const double y2r = fma(h2, k1r, zr), y2i = fma(h2, k1i, zi);
        const double m2  = a - fma(y2r, y2r, y2i * y2i);
        const double k2r = m2 * y2r, k2i = m2 * y2i;

        const double y3r = fma(h2, k2r, zr), y3i = fma(h2, k2i, zi);
        const double m3  = a - fma(y3r, y3r, y3i * y3i);
        const double k3r = m3 * y3r, k3i = m3 * y3i;

        const double y4r = fma(h, k3r, zr), y4i = fma(h, k3i, zi);
        const double m4  = a - fma(y4r, y4r, y4i * y4i);
        const double k4r = m4 * y4r, k4i = m4 * y4i;

        zr = fma(h6, k1r + 2.0 * (k2r + k3r) + k4r, zr);
        zi = fma(h6, k1i + 2.0 * (k2i + k3i) + k4i, zi);
    }

    zT[2 * i + 0] = zr;
    zT[2 * i + 1] = zi;
}

// ---------------------------------------------------------------------------
// inputs (setup_inputs order): x f32[4096], U c128[4096^2] (DEAD - skipped),
//                              W c128[4096^2], z c128[4096]
// output: z(T) complex128 -> interleaved doubles in d_out.
// ---------------------------------------------------------------------------
extern "C" void kernel_launch(void* const* d_in, const int* in_sizes, int n_in,
                              void* d_out, int out_size, void* d_ws, size_t ws_size,
                              hipStream_t stream) {
    const float*   x = (const float*)d_in[0];
    // d_in[1] = U : unused (U @ z result is dead in the reference)
    const double2* W = (const double2*)d_in[2];
    const double*  z = (const double*)d_in[3];

    double* a = (double*)d_ws;   // 4096 doubles = 32 KB scratch

    rewx_matvec<<<NROW / ROWS_PER_BLOCK, 256, 0, stream>>>(W, x, a);
    hopf_rk4<<<NROW / 256, 256, 0, stream>>>(a, z, (double*)d_out);
}